// GraphConvolution_5909875000109
// MI455X (gfx1250) — compile-verified
//
#include <hip/hip_runtime.h>
#include <cstdint>

// CDNA5 fp32 WMMA fragment types: 16x16x4 f32 -> A/B are 2 floats/lane, C/D are 8 floats/lane.
typedef __attribute__((ext_vector_type(2))) float v2f;
typedef __attribute__((ext_vector_type(8))) float v8f;

#define S_NEIGH 10
#define F_DIM   128
#define ROW_PAD 132            // 128 + 4 -> LDS bank-conflict-free strided access
#define TILE_M  16             // rows of output per workgroup
#define NWAVES  8              // 256 threads / wave32

__global__ __launch_bounds__(256)
void gcn_fused_kernel(const int* __restrict__ idx,
                      const float* __restrict__ emb,
                      const float* __restrict__ w,
                      float* __restrict__ out)
{
    // 64 KB weight (padded) + 8 KB aggregated tile (padded) ~ 76 KB of the 320 KB WGP LDS
    __shared__ float ldsW[F_DIM * ROW_PAD];   // weight[n][k], row-padded
    __shared__ float ldsA[TILE_M * ROW_PAD];  // aggregated features, row-padded

    const int tid     = threadIdx.x;
    const int wave    = tid >> 5;      // 0..7
    const int lane    = tid & 31;
    const int rowBase = blockIdx.x * TILE_M;

    // ---- Stage 1: cooperative copy of weight [128][128] into padded LDS ----
    // 2 threads per weight row, 64 floats (16 float4) each; both src and dst 16B-aligned.
    {
        const int wr   = tid >> 1;        // weight row 0..127
        const int half = tid & 1;         // which 64-float half of the row
        const float4* src = (const float4*)(w + wr * F_DIM + half * 64);
        float4*       dst = (float4*)(&ldsW[wr * ROW_PAD + half * 64]);
        #pragma unroll
        for (int i = 0; i < 16; ++i) dst[i] = src[i];
    }

    // ---- Stage 2: gather + mean. One wave per output row; each wave does 2 rows. ----
    // 32 lanes x float4 = one full 512B embedding row per load -> perfectly coalesced.
    const float scale = (float)(1.0 / (10.0 + 1e-10));
    #pragma unroll
    for (int rr = 0; rr < 2; ++rr) {
        const int r   = wave + rr * NWAVES;      // 0..15
        const int row = rowBase + r;
        float4 acc = make_float4(0.f, 0.f, 0.f, 0.f);
        #pragma unroll
        for (int s = 0; s < S_NEIGH; ++s) {
            const int nb = idx[row * S_NEIGH + s];
            const float4 v = ((const float4*)(emb + (int64_t)nb * F_DIM))[lane];
            acc.x += v.x; acc.y += v.y; acc.z += v.z; acc.w += v.w;
        }
        acc.x *= scale; acc.y *= scale; acc.z *= scale; acc.w *= scale;
        ((float4*)(&ldsA[r * ROW_PAD]))[lane] = acc;
    }
    __syncthreads();

    // ---- Stage 3: WMMA GEMM. Wave w computes out[rowBase:+16, w*16:+16]. ----
    // out[m][n] = sum_k agg[m][k] * weight[n][k]   (B fragment == weight row n)
    const int nBase = wave * TILE_M;
    const int half  = lane >> 4;        // 0: K pair {0,1}, 1: K pair {2,3}
    const int l     = lane & 15;
    const int koff  = half * 2;
    const float* aRow = &ldsA[l * ROW_PAD + koff];            // A: M = l
    const float* bRow = &ldsW[(nBase + l) * ROW_PAD + koff];  // B: N = l

    v8f c = {};
    #pragma unroll
    for (int k = 0; k < F_DIM; k += 4) {
        v2f a, b;
        a.x = aRow[k];  a.y = aRow[k + 1];   // 8B-aligned pair -> ds_load_b64
        b.x = bRow[k];  b.y = bRow[k + 1];
        c = __builtin_amdgcn_wmma_f32_16x16x4_f32(
                /*neg_a=*/false, a, /*neg_b=*/false, b,
                /*c_mod=*/(short)0, c, /*reuse_a=*/false, /*reuse_b=*/false);
    }

    // ---- Stage 4: ReLU + store. D layout: VGPR r -> M = r + 8*half, N = l. ----
    const int col = nBase + l;
    #pragma unroll
    for (int r = 0; r < 8; ++r) {
        const int row = rowBase + r + half * 8;
        out[row * F_DIM + col] = fmaxf(c[r], 0.0f);
    }
}

extern "C" void kernel_launch(void* const* d_in, const int* in_sizes, int n_in,
                              void* d_out, int out_size, void* d_ws, size_t ws_size,
                              hipStream_t stream) {
    const int*   idx = (const int*)d_in[0];    // neighbor_idx [B,10] (int32 on-device)
    const float* emb = (const float*)d_in[1];  // embeddings [1e6,128] f32
    const float* w   = (const float*)d_in[2];  // weight [128,128] f32
    float*       out = (float*)d_out;          // [B,128] f32

    const int B = in_sizes[0] / S_NEIGH;       // 16384
    dim3 grid(B / TILE_M), block(256);
    hipLaunchKernelGGL(gcn_fused_kernel, grid, block, 0, stream, idx, emb, w, out);
}